// Detector_12541304505002
// MI455X (gfx1250) — compile-verified
//
#include <hip/hip_runtime.h>
#include <math.h>

#define NUM_CLS   80
#define CH        85
#define TOPK      100
#define HW_TOTAL  5456
#define BATCH_N   64
#define CONF_TH   0.05f
#define NMS_TH    0.5f
#define PADN      112      // 7 * 16 (WMMA tile padding of 100)
#define NFLAT     (NUM_CLS * TOPK)   // 8000

typedef float v2f __attribute__((ext_vector_type(2)));
typedef float v8f __attribute__((ext_vector_type(8)));

// ---- monotone float <-> uint key helpers (stable top-k tie-breaking) ----
__device__ __forceinline__ unsigned f2key(float v) {
  unsigned b = __float_as_uint(v);
  return (b & 0x80000000u) ? ~b : (b | 0x80000000u);
}
__device__ __forceinline__ float key2f(unsigned k) {
  unsigned b = (k & 0x80000000u) ? (k ^ 0x80000000u) : ~k;
  return __uint_as_float(b);
}

// =====================================================================
// Kernel 1: per-(batch,class) top-100 over HW=5456 positions.
// Scores staged to LDS via CDNA5 async global->LDS loads when available.
// =====================================================================
__global__ __launch_bounds__(256) void k_topk_class(
    const float* __restrict__ pred,
    float* __restrict__ cand_score, int* __restrict__ cand_hw)
{
  const int c = blockIdx.x;       // class
  const int b = blockIdx.y;       // batch
  const int t = threadIdx.x;

  __shared__ unsigned long long key[HW_TOTAL];   // 43648 B
  __shared__ float stage[2048];                  //  8192 B
  __shared__ unsigned long long red[256];        //  2048 B

  const size_t rowbase = (size_t)b * HW_TOTAL;

  for (int base = 0; base < HW_TOTAL; base += 2048) {
    int n = HW_TOTAL - base; if (n > 2048) n = 2048;
#if defined(__gfx1250__) && __has_builtin(__builtin_amdgcn_global_load_async_to_lds_b32)
    for (int i = t; i < n; i += 256) {
      const float* g = pred + (rowbase + base + i) * CH + c;
      __builtin_amdgcn_global_load_async_to_lds_b32(
          (__attribute__((address_space(1))) int*)g,
          (__attribute__((address_space(3))) int*)&stage[i],
          0, 0);
    }
#if __has_builtin(__builtin_amdgcn_s_wait_asynccnt)
    __builtin_amdgcn_s_wait_asynccnt(0);
#else
    asm volatile("s_wait_asynccnt 0" ::: "memory");
#endif
#else
    for (int i = t; i < n; i += 256)
      stage[i] = pred[(rowbase + base + i) * CH + c];
#endif
    __syncthreads();
    for (int i = t; i < n; i += 256) {
      int hw = base + i;
      float raw = stage[i];
      float nk  = pred[(rowbase + hw) * CH + (CH - 1)];   // L2-resident gather
      float s = 1.0f / (1.0f + __expf(-raw));             // sigmoid(cls)
      float q = 1.0f / (1.0f + __expf(-nk));              // sigmoid(nks)
      float v = __powf(s, 2.0f - q);                      // recalibrated score
      key[hw] = ((unsigned long long)f2key(v) << 32) | (unsigned)~hw;
    }
    __syncthreads();
  }

  // 100 rounds of workgroup argmax over LDS keys
  const size_t obase = ((size_t)b * NUM_CLS + c) * TOPK;
  for (int r = 0; r < TOPK; ++r) {
    unsigned long long best = 0ull;
    for (int hw = t; hw < HW_TOTAL; hw += 256) {
      unsigned long long k = key[hw];
      best = (k > best) ? k : best;
    }
    red[t] = best;
    __syncthreads();
    for (int s = 128; s > 0; s >>= 1) {
      if (t < s && red[t + s] > red[t]) red[t] = red[t + s];
      __syncthreads();
    }
    if (t == 0) {
      unsigned long long win = red[0];
      int hw = (int)(~(unsigned)win);
      cand_score[obase + r] = key2f((unsigned)(win >> 32));
      cand_hw[obase + r]    = hw;
      key[hw] = 0ull;   // remove winner (real keys all have MSB set)
    }
    __syncthreads();
  }
}

// =====================================================================
// Kernel 2: per-image stable top-100 over 8000 (class,rank) candidates.
// =====================================================================
__global__ __launch_bounds__(256) void k_merge(
    const float* __restrict__ cand_score, const int* __restrict__ cand_hw,
    float* __restrict__ sel_score, int* __restrict__ sel_cls,
    int* __restrict__ sel_hw,
    float* __restrict__ out_scores, int* __restrict__ out_cls)
{
  const int b = blockIdx.x;
  const int t = threadIdx.x;
  __shared__ float sc[NFLAT];                 // 32000 B
  __shared__ unsigned long long red[256];

  for (int f = t; f < NFLAT; f += 256) sc[f] = cand_score[(size_t)b * NFLAT + f];
  __syncthreads();

  for (int r = 0; r < TOPK; ++r) {
    unsigned long long best = 0ull;
    for (int f = t; f < NFLAT; f += 256) {
      unsigned long long k = ((unsigned long long)f2key(sc[f]) << 32) | (unsigned)~f;
      best = (k > best) ? k : best;
    }
    red[t] = best;
    __syncthreads();
    for (int s = 128; s > 0; s >>= 1) {
      if (t < s && red[t + s] > red[t]) red[t] = red[t + s];
      __syncthreads();
    }
    if (t == 0) {
      int f = (int)(~(unsigned)red[0]);
      float v = sc[f];
      sel_score[b * TOPK + r]  = v;
      sel_cls[b * TOPK + r]    = f / TOPK;
      sel_hw[b * TOPK + r]     = cand_hw[(size_t)b * NFLAT + f];
      out_scores[b * TOPK + r] = v;
      out_cls[b * TOPK + r]    = f / TOPK;
      sc[f] = -INFINITY;
    }
    __syncthreads();
  }
}

// =====================================================================
// Kernel 3: box decode + class-aware NMS + clip/normalize.
// areas[i]+areas[j] outer-sum computed on the matrix pipe (WMMA f32 16x16x4).
// =====================================================================
__global__ __launch_bounds__(256) void k_nms(
    const float* __restrict__ pred, const float* __restrict__ ploc,
    const float* __restrict__ sel_score, const int* __restrict__ sel_cls,
    const int* __restrict__ sel_hw,
    float* __restrict__ out_boxes, int* __restrict__ out_keep)
{
  const int b = blockIdx.x;
  const int t = threadIdx.x;
  __shared__ float X1[PADN], Y1[PADN], X2[PADN], Y2[PADN], AR[PADN];
  __shared__ int   CL[PADN], KP[PADN];
  __shared__ unsigned char SUP[PADN * PADN];   // 12544 B

  // ---- decode the 100 selected boxes ----
  if (t < PADN) {
    float x1 = 0.f, y1 = 0.f, x2 = 0.f, y2 = 0.f, a = 0.f;
    int cl = -1, kp = 0;
    if (t < TOPK) {
      int hw = sel_hw[b * TOPK + t];
      const float* p = pred + ((size_t)b * HW_TOTAL + hw) * CH + NUM_CLS;
      float e0 = __expf(p[0]), e1 = __expf(p[1]);
      float e2 = __expf(p[2]), e3 = __expf(p[3]);
      x1 = ploc[hw * 4 + 0] - e0;
      y1 = ploc[hw * 4 + 1] - e1;
      x2 = ploc[hw * 4 + 2] + e2;
      y2 = ploc[hw * 4 + 3] + e3;
      a  = (x2 - x1) * (y2 - y1);
      cl = sel_cls[b * TOPK + t];
      kp = (sel_score[b * TOPK + t] >= CONF_TH) ? 1 : 0;
    }
    X1[t] = x1; Y1[t] = y1; X2[t] = x2; Y2[t] = y2;
    AR[t] = a;  CL[t] = cl; KP[t] = kp;
  }
  __syncthreads();

  // ---- suppression matrix: 7x7 tiles of 16x16, one tile per wave ----
  const int wave = t >> 5;
  const int lane = t & 31;
  for (int tt = wave; tt < 49; tt += 8) {
    const int i0 = (tt / 7) * 16;
    const int j0 = (tt % 7) * 16;
    const int nn = j0 + (lane & 15);
    const int mbase = i0 + ((lane >> 4) << 3);
    v8f D;
#if defined(__gfx1250__) && __has_builtin(__builtin_amdgcn_wmma_f32_16x16x4_f32)
    {
      // A (16x4): col0 = areas_i, col1 = 1;  B (4x16): row0 = 1, row1 = areas_j
      // => D[m][n] = areas[i0+m] + areas[j0+n]
      const int l16 = lane & 15;
      const bool lo = lane < 16;        // lanes 16..31 hold K=2,3 -> zero
      v2f A, B;
      A.x = lo ? AR[i0 + l16] : 0.0f;
      A.y = lo ? 1.0f         : 0.0f;
      B.x = lo ? 1.0f         : 0.0f;
      B.y = lo ? AR[j0 + l16] : 0.0f;
      v8f Cz = {};
      D = __builtin_amdgcn_wmma_f32_16x16x4_f32(false, A, false, B,
                                                (short)0, Cz, false, false);
    }
#else
    for (int r = 0; r < 8; ++r) D[r] = AR[mbase + r] + AR[nn];
#endif
    for (int r = 0; r < 8; ++r) {
      int i = mbase + r;
      float xx1 = fmaxf(X1[i], X1[nn]);
      float yy1 = fmaxf(Y1[i], Y1[nn]);
      float xx2 = fminf(X2[i], X2[nn]);
      float yy2 = fminf(Y2[i], Y2[nn]);
      float w = fmaxf(1e-28f, xx2 - xx1);
      float h = fmaxf(1e-28f, yy2 - yy1);
      float inter = w * h;
      float iou = inter / (D[r] - inter);   // D[r] = area_i + area_j
      SUP[i * PADN + nn] =
          (iou > NMS_TH) && (CL[i] == CL[nn]) && (nn > i) &&
          (i < TOPK) && (nn < TOPK);
    }
  }
  __syncthreads();

  // ---- sequential greedy suppression propagation (100 steps) ----
  for (int i = 0; i < TOPK; ++i) {
    int ki = KP[i];                               // uniform read
    if (t < TOPK && ki && SUP[i * PADN + t]) KP[t] = 0;  // writes only t > i
    __syncthreads();
  }

  // ---- clip, normalize, emit ----
  if (t < TOPK) {
    const float inv = 1.0f / 512.0f;
    float bx0 = fminf(fmaxf(X1[t], 0.0f), 511.0f) * inv;
    float bx1 = fminf(fmaxf(Y1[t], 0.0f), 511.0f) * inv;
    float bx2 = fminf(fmaxf(X2[t], 0.0f), 511.0f) * inv;
    float bx3 = fminf(fmaxf(Y2[t], 0.0f), 511.0f) * inv;
    size_t o = ((size_t)b * TOPK + t) * 4;
    out_boxes[o + 0] = bx0;
    out_boxes[o + 1] = bx1;
    out_boxes[o + 2] = bx2;
    out_boxes[o + 3] = bx3;
    out_keep[b * TOPK + t] = KP[t];
  }
}

// =====================================================================
extern "C" void kernel_launch(void* const* d_in, const int* in_sizes, int n_in,
                              void* d_out, int out_size, void* d_ws, size_t ws_size,
                              hipStream_t stream)
{
  const float* pred = (const float*)d_in[0];   // (64, 5456, 85)
  const float* ploc = (const float*)d_in[1];   // (5456, 4)

  char* ws = (char*)d_ws;
  float* cand_score = (float*)(ws + 0);          // 64*8000 f32 = 2,048,000 B
  int*   cand_hw    = (int*)  (ws + 2048000);    // 64*8000 i32 = 2,048,000 B
  int*   sel_hw     = (int*)  (ws + 4096000);    // 64*100 i32
  int*   sel_cls    = (int*)  (ws + 4121600);    // 64*100 i32
  float* sel_score  = (float*)(ws + 4147200);    // 64*100 f32

  float* out        = (float*)d_out;
  float* out_boxes  = out;                       // (64,100,4) f32
  float* out_scores = out + 25600;               // (64,100)   f32
  int*   out_cls    = (int*)(out + 32000);       // (64,100)   i32 bits
  int*   out_keep   = (int*)(out + 38400);       // (64,100)   0/1

  dim3 g1(NUM_CLS, BATCH_N);
  k_topk_class<<<g1, 256, 0, stream>>>(pred, cand_score, cand_hw);
  k_merge<<<BATCH_N, 256, 0, stream>>>(cand_score, cand_hw,
                                       sel_score, sel_cls, sel_hw,
                                       out_scores, out_cls);
  k_nms<<<BATCH_N, 256, 0, stream>>>(pred, ploc, sel_score, sel_cls, sel_hw,
                                     out_boxes, out_keep);
}